// LogicConv3d_85504208929322
// MI455X (gfx1250) — compile-verified
//
#include <hip/hip_runtime.h>
#include <hip/hip_bf16.h>
#include <stdint.h>

// Problem constants (from reference)
#define NB 16
#define NC 3
#define NH 32
#define NW 32
#define NK 64
#define NP 784          // 28*28 patches
#define NS 64           // leaves per tree
#define NNODE 127       // 64+32+16+8+4+2+1 gate nodes per (k)-tree
#define NROWS (NNODE * NK)   // 8128 (node,k) rows
#define XELEMS (NC * NH * NW) // 3072 floats = 12KB per batch image

typedef __attribute__((ext_vector_type(2))) float        v2f;
typedef __attribute__((ext_vector_type(8))) float        v8f;
typedef __attribute__((ext_vector_type(4))) unsigned int v4u;
typedef __attribute__((ext_vector_type(8))) int          v8i;
typedef __attribute__((ext_vector_type(4))) int          v4i;

__device__ __constant__ float GC[16][4] = {
  {0,0,0,0},{0,0,0,1},{0,1,0,-1},{0,1,0,0},
  {0,0,1,-1},{0,0,1,0},{0,1,1,-2},{0,1,1,-1},
  {1,-1,-1,1},{1,-1,-1,2},{1,0,-1,0},{1,0,-1,1},
  {1,-1,0,0},{1,-1,0,1},{1,0,0,-1},{1,0,0,0}};

// ---------------------------------------------------------------------------
// Kernel 1: coef = softmax(logits) @ GATE_COEFFS  via V_WMMA_F32_16X16X4_F32
// rows = node*64 + k  (8128 rows, inner dim 16, 4 output coefs padded to 16)
// Output layout: cw[(k*127 + node)*4 + j]
// ---------------------------------------------------------------------------
__global__ __launch_bounds__(256) void coef_kernel(
    const float* __restrict__ w0, const float* __restrict__ w1,
    const float* __restrict__ w2, const float* __restrict__ w3,
    const float* __restrict__ w4, const float* __restrict__ w5,
    const float* __restrict__ w6, float* __restrict__ cw) {
  __shared__ float probs[128][17];   // +1 pad to dodge bank conflicts
  __shared__ float gcs[16][16];      // GATE_COEFFS zero-padded to 16 cols
  const int tid  = threadIdx.x;
  const int row0 = blockIdx.x * 128;

  // Stage B-matrix source (16x4 table, zero-padded to 16x16) into LDS so the
  // WMMA B feed is a branch-free ds_load (EXEC stays all-ones throughout).
  {
    const int r = tid >> 4, c = tid & 15;   // 256 threads cover 16x16
    gcs[r][c] = (c < 4) ? GC[r][c] : 0.0f;
  }

  // Phase 1: threads 0..127 each softmax one (node,k) row of 16 logits.
  if (tid < 128) {
    const int row = row0 + tid;
    if (row < NROWS) {
      const int node = row >> 6;
      const int k    = row & 63;
      const float* wp; int s;
      if      (node <  64) { wp = w0; s = node;       }
      else if (node <  96) { wp = w1; s = node - 64;  }
      else if (node < 112) { wp = w2; s = node - 96;  }
      else if (node < 120) { wp = w3; s = node - 112; }
      else if (node < 124) { wp = w4; s = node - 120; }
      else if (node < 126) { wp = w5; s = node - 124; }
      else                 { wp = w6; s = 0;          }
      const float* lp = wp + ((size_t)s * NK + k) * 16;
      float v[16]; float mx = -3.0e38f;
      #pragma unroll
      for (int i = 0; i < 16; ++i) { v[i] = lp[i]; mx = fmaxf(mx, v[i]); }
      float sum = 0.0f;
      #pragma unroll
      for (int i = 0; i < 16; ++i) { v[i] = __expf(v[i] - mx); sum += v[i]; }
      const float inv = 1.0f / sum;
      #pragma unroll
      for (int i = 0; i < 16; ++i) probs[tid][i] = v[i] * inv;
    } else {
      #pragma unroll
      for (int i = 0; i < 16; ++i) probs[tid][i] = 0.0f;
    }
  }
  __syncthreads();

  // Phase 2: each wave multiplies its 16 rows by G (16x4, padded to 16 cols)
  // accumulating four K=4 chunks with v_wmma_f32_16x16x4_f32.
  const int lane = tid & 31;
  const int wave = tid >> 5;
  const int half = lane >> 4;   // A/B: K = 2*half + j   (per ISA 16x4 layout)
  const int m    = lane & 15;   // A row / D col
  const int rloc = wave * 16;

  v8f acc = {0.f,0.f,0.f,0.f,0.f,0.f,0.f,0.f};
  #pragma unroll
  for (int t = 0; t < 4; ++t) {
    v2f A, Bv;
    A[0]  = probs[rloc + m][4*t + 2*half + 0];
    A[1]  = probs[rloc + m][4*t + 2*half + 1];
    Bv[0] = gcs[4*t + 2*half + 0][m];
    Bv[1] = gcs[4*t + 2*half + 1][m];
    acc = __builtin_amdgcn_wmma_f32_16x16x4_f32(
        /*neg_a=*/false, A, /*neg_b=*/false, Bv,
        /*c_mod=*/(short)0, acc, /*reuse_a=*/false, /*reuse_b=*/false);
  }

  // D layout: vgpr v, lanes0-15 -> M=v, lanes16-31 -> M=v+8; N = lane%16.
  if (m < 4) {
    #pragma unroll
    for (int v = 0; v < 8; ++v) {
      const int row = row0 + rloc + v + 8*half;
      if (row < NROWS) {
        const int node = row >> 6;
        const int k    = row & 63;
        cw[((size_t)k * NNODE + node) * 4 + m] = acc[v];
      }
    }
  }
}

// ---------------------------------------------------------------------------
// Kernel 2: one workgroup per (b,k). TDM-stage x[b] (12KB) into LDS, then each
// thread runs whole trees: 64 LDS gathers/side + 127 fused gate ops, all in
// registers. Index streams are contiguous 768B/thread -> b128 loads.
// ---------------------------------------------------------------------------
__global__ __launch_bounds__(256) void logic_tree_kernel(
    const float* __restrict__ x, const int* __restrict__ lidx,
    const int* __restrict__ ridx, const float* __restrict__ cw,
    float* __restrict__ out) {
  __shared__ float  xs[XELEMS];     // x[b], 12KB
  __shared__ float4 cf[NNODE];      // 127 coefficient quads for this k
  const int tid = threadIdx.x;
  const int k   = blockIdx.x;
  const int b   = blockIdx.y;

  // --- Tensor Data Mover: 1-D copy of 3072 f32 from global to LDS ---------
  if (tid == 0) {
    const uint64_t ga = (uint64_t)(uintptr_t)(x + (size_t)b * XELEMS);
    const uint32_t lo = (uint32_t)(uintptr_t)(&xs[0]);   // LDS byte offset
    v4u g0 = {0u, 0u, 0u, 0u};
    g0[0] = 1u;                                      // count=1 (valid D#)
    g0[1] = lo;                                      // lds_addr
    g0[2] = (uint32_t)ga;                            // global_addr[31:0]
    g0[3] = (uint32_t)((ga >> 32) & 0x01FFFFFFu) | 0x80000000u; // [56:32]|type=2
    v8i g1 = {0,0,0,0,0,0,0,0};
    g1[0] = 0x00020000;                              // data_size=4B
    g1[1] = (XELEMS & 0xFFFF) << 16;                 // tensor_dim0 lo16
    g1[2] = ((XELEMS >> 16) & 0xFFFF) | (1 << 16);   // dim0 hi | tensor_dim1=1
    g1[3] = (XELEMS & 0xFFFF) << 16;                 // tile_dim0 = 3072
    g1[5] = XELEMS;                                  // tensor_dim0_stride
    v4i gz4 = {0,0,0,0};
    v8i gz8 = {0,0,0,0,0,0,0,0};
    // clang-23 / therock-10.0 form: 6 args (g0, g1, g2, g3, g4, cpol)
    __builtin_amdgcn_tensor_load_to_lds(g0, g1, gz4, gz4, gz8, 0);
    __builtin_amdgcn_s_wait_tensorcnt(0);
  }
  // coefficient quads for this k (broadcast-friendly in LDS)
  const float4* cwsrc = (const float4*)cw + (size_t)k * NNODE;
  for (int i = tid; i < NNODE; i += 256) cf[i] = cwsrc[i];
  __syncthreads();

  #pragma unroll 1
  for (int it = 0; it < 4; ++it) {
    const int p = it * 256 + tid;
    if (p < NP) {
      float cur[NS];
      const int4* l4 = (const int4*)lidx + (size_t)(k * NP + p) * 48; // 768B/16
      const int4* r4 = (const int4*)ridx + (size_t)(k * NP + p) * 48;

      // Level 0: 64 leaves, 4 per group of three b128 index loads.
      // idx triplet = (h, w, c); flat = (c<<10) + (h<<5) + w
      #pragma unroll
      for (int g = 0; g < 16; ++g) {
        const int4 L0 = l4[3*g+0], L1 = l4[3*g+1], L2 = l4[3*g+2];
        const int4 R0 = r4[3*g+0], R1 = r4[3*g+1], R2 = r4[3*g+2];
        const float a0 = xs[(L0.z<<10)+(L0.x<<5)+L0.y];
        const float b0 = xs[(R0.z<<10)+(R0.x<<5)+R0.y];
        const float a1 = xs[(L1.y<<10)+(L0.w<<5)+L1.x];
        const float b1 = xs[(R1.y<<10)+(R0.w<<5)+R1.x];
        const float a2 = xs[(L2.x<<10)+(L1.z<<5)+L1.w];
        const float b2 = xs[(R2.x<<10)+(R1.z<<5)+R1.w];
        const float a3 = xs[(L2.w<<10)+(L2.y<<5)+L2.z];
        const float b3 = xs[(R2.w<<10)+(R2.y<<5)+R2.z];
        const float4 c0 = cf[4*g+0], c1 = cf[4*g+1];
        const float4 c2 = cf[4*g+2], c3 = cf[4*g+3];
        cur[4*g+0] = fmaf(c0.w, a0*b0, fmaf(c0.z, b0, fmaf(c0.y, a0, c0.x)));
        cur[4*g+1] = fmaf(c1.w, a1*b1, fmaf(c1.z, b1, fmaf(c1.y, a1, c1.x)));
        cur[4*g+2] = fmaf(c2.w, a2*b2, fmaf(c2.z, b2, fmaf(c2.y, a2, c2.x)));
        cur[4*g+3] = fmaf(c3.w, a3*b3, fmaf(c3.z, b3, fmaf(c3.y, a3, c3.x)));
      }

      // Levels 1..6: a = cur[2s], b = cur[2s+1]; node base 64,96,112,120,124,126
      #define DO_LEVEL(CNT, BASE)                                            \
        _Pragma("unroll")                                                    \
        for (int s = 0; s < (CNT); ++s) {                                    \
          const float av = cur[2*s], bv = cur[2*s+1];                        \
          const float4 c = cf[(BASE) + s];                                   \
          cur[s] = fmaf(c.w, av*bv, fmaf(c.z, bv, fmaf(c.y, av, c.x)));      \
        }
      DO_LEVEL(32,  64)
      DO_LEVEL(16,  96)
      DO_LEVEL( 8, 112)
      DO_LEVEL( 4, 120)
      DO_LEVEL( 2, 124)
      DO_LEVEL( 1, 126)
      #undef DO_LEVEL

      out[((size_t)b * NK + k) * NP + p] = cur[0];
    }
  }
}

// ---------------------------------------------------------------------------
extern "C" void kernel_launch(void* const* d_in, const int* in_sizes, int n_in,
                              void* d_out, int out_size, void* d_ws, size_t ws_size,
                              hipStream_t stream) {
  (void)in_sizes; (void)n_in; (void)out_size; (void)ws_size;
  const float* x  = (const float*)d_in[0];
  const float* w0 = (const float*)d_in[1];
  const float* w1 = (const float*)d_in[2];
  const float* w2 = (const float*)d_in[3];
  const float* w3 = (const float*)d_in[4];
  const float* w4 = (const float*)d_in[5];
  const float* w5 = (const float*)d_in[6];
  const float* w6 = (const float*)d_in[7];
  const int*   li = (const int*)d_in[8];
  const int*   ri = (const int*)d_in[9];
  float* outp = (float*)d_out;
  float* cw   = (float*)d_ws;    // 8128*4 f32 = 130KB of scratch

  // 64 blocks x 128 rows covers 8128 (node,k) rows.
  coef_kernel<<<64, 256, 0, stream>>>(w0, w1, w2, w3, w4, w5, w6, cw);

  dim3 grid(NK, NB);             // one workgroup per (b,k)
  logic_tree_kernel<<<grid, 256, 0, stream>>>(x, li, ri, cw, outp);
}